// GeToInformedNeighborSampler_13675175870556
// MI455X (gfx1250) — compile-verified
//
#include <hip/hip_runtime.h>
#include <hip/hip_bf16.h>

typedef __attribute__((ext_vector_type(2))) float v2f;
typedef __attribute__((ext_vector_type(8))) float v8f;

#define NUM_NODES 20000
#define MAX_DEG   4096
#define B         256
#define S         16
#define GD        128
#define NSAMP     4096   // B*S == total_num_adj_samples == MAX_DEG here

// ---------------------------------------------------------------------------
// Kernel 1: logits[c] = log(1 / ||target - geto_elms[node_samples[c]]||)
// for c = b*16 + s, node_samples[c] = geto_adj_info[geto_ids[b], s].
// Uses V_WMMA_F32_16X16X4_F32 for the t·h dot products (f32 matrix path),
// VALU for the squared norms:  d2 = ||t||^2 + ||h||^2 - 2 t·h.
// One wave per 16-row tile; 8 waves/block, 32 blocks -> 256 tiles (= batches).
// ---------------------------------------------------------------------------
__global__ __launch_bounds__(256)
void geto_logits_kernel(const int* __restrict__ geto_adj_info,
                        const float* __restrict__ geto_elms,
                        const int* __restrict__ ids,
                        const int* __restrict__ geto_ids,
                        float* __restrict__ logits) {
    __shared__ float sdot[8 * 16];  // per-wave 16 dot products

    const int lane = threadIdx.x & 31;
    const int wave = threadIdx.x >> 5;
    const int tile = blockIdx.x * 8 + wave;   // == batch index b, 0..255

    const int m = lane & 15;   // row within tile (M)
    const int h = lane >> 4;   // half: selects K pair {2h, 2h+1} of each K4 chunk

    // node_samples[tile*16 + m] = geto_adj_info[geto_ids[tile], m]
    const int node = geto_adj_info[(size_t)geto_ids[tile] * MAX_DEG + m];
    const float* __restrict__ hrow = geto_elms + (size_t)node * GD;
    const float* __restrict__ trow = geto_elms + (size_t)ids[0] * GD;

    v8f acc = {};          // C/D accumulator: 16x16 f32
    float nh = 0.0f;       // partial ||h_m||^2  (this lane's K subset)
    float nt = 0.0f;       // partial ||t||^2

    // f32 A-matrix 16x4 layout: lanes 0-15 hold (K=0,K=1), lanes 16-31 (K=2,K=3).
    // B 4x16 mirrors it; every column of B is the target chunk.
#pragma unroll 8
    for (int kc = 0; kc < GD / 4; ++kc) {
        const int kb = 4 * kc + 2 * h;
        v2f a, b;
        a.x = hrow[kb]; a.y = hrow[kb + 1];
        b.x = trow[kb]; b.y = trow[kb + 1];
        nh = fmaf(a.x, a.x, fmaf(a.y, a.y, nh));
        nt = fmaf(b.x, b.x, fmaf(b.y, b.y, nt));
        // 8 args: (neg_a, A, neg_b, B, c_mod, C, reuse_a, reuse_b)
        acc = __builtin_amdgcn_wmma_f32_16x16x4_f32(
            false, a, false, b, (short)0, acc, false, false);
    }

    // Combine the two K-halves (lane l <-> lane l^16) of the norms.
    nh += __shfl_xor(nh, 16);
    nt += __shfl_xor(nt, 16);

    // C layout: VGPR r, lanes 0-15 -> M=r (N=lane), lanes 16-31 -> M=r+8.
    // Every column holds the same dot; lanes with N==0 spill their 8 rows.
    if (m == 0) {
        const int base = wave * 16 + h * 8;
        sdot[base + 0] = acc[0]; sdot[base + 1] = acc[1];
        sdot[base + 2] = acc[2]; sdot[base + 3] = acc[3];
        sdot[base + 4] = acc[4]; sdot[base + 5] = acc[5];
        sdot[base + 6] = acc[6]; sdot[base + 7] = acc[7];
    }
    __syncthreads();

    if (h == 0) {  // lanes 0..15: one row each
        const float dot  = sdot[wave * 16 + m];
        const float d2   = fmaxf(nt + nh - 2.0f * dot, 0.0f);
        const float dist = sqrtf(d2);
        const float w    = 1.0f / dist;          // reference: 1/dist then log
        logits[tile * 16 + m] = logf(w);
    }
}

// ---------------------------------------------------------------------------
// threefry2x32 (JAX key schedule), returning x0 of the output pair.
// key = threefry_seed(42) = (0, 42).
// ---------------------------------------------------------------------------
__device__ __forceinline__ unsigned int rotl32(unsigned int x, int r) {
    return (x << r) | (x >> (32 - r));
}

__device__ __forceinline__ unsigned int threefry2x32_x0(unsigned int k0,
                                                        unsigned int k1,
                                                        unsigned int x0,
                                                        unsigned int x1) {
    const unsigned int k2 = k0 ^ k1 ^ 0x1BD11BDAu;
    x0 += k0; x1 += k1;
#define TF_R4(a, b, c, d) \
    x0 += x1; x1 = rotl32(x1, a); x1 ^= x0; \
    x0 += x1; x1 = rotl32(x1, b); x1 ^= x0; \
    x0 += x1; x1 = rotl32(x1, c); x1 ^= x0; \
    x0 += x1; x1 = rotl32(x1, d); x1 ^= x0;
    TF_R4(13, 15, 26, 6)  x0 += k1; x1 += k2 + 1u;
    TF_R4(17, 29, 16, 24) x0 += k2; x1 += k0 + 2u;
    TF_R4(13, 15, 26, 6)  x0 += k0; x1 += k1 + 3u;
    TF_R4(17, 29, 16, 24) x0 += k1; x1 += k2 + 4u;
    TF_R4(13, 15, 26, 6)  x0 += k2;
#undef TF_R4
    return x0;
}

// ---------------------------------------------------------------------------
// Kernel 2: one block per draw j (0..15).  idx[j] = argmax_c(logits[c] + g_jc)
// with JAX Gumbel bits: flat i = j*4096 + c < 2^23  =>
//   bits = threefry2x32((0,42), (i, i + 2^23)).x0
//   u    = max(tiny, bitcast((bits>>9)|0x3f800000) - 1 + tiny)
//   g    = -log(-log(u))
// Tie-break: lowest index (jnp.argmax).  Then the same block gathers the two
// (256,) output columns for its j.
// ---------------------------------------------------------------------------
__global__ __launch_bounds__(256)
void categorical_gather_kernel(const int* __restrict__ adj_info,
                               const int* __restrict__ geto_adj_info,
                               const int* __restrict__ ids,
                               const int* __restrict__ geto_ids,
                               const float* __restrict__ logits,
                               float* __restrict__ out) {
    __shared__ float sval[256];
    __shared__ int   sidx[256];

    const int j = blockIdx.x;      // draw index, 0..15
    const int t = threadIdx.x;     // 0..255
    const float tiny = 1.17549435e-38f;

    float best = -__builtin_inff();
    int   bi   = 0x7fffffff;

#pragma unroll 4
    for (int r = 0; r < NSAMP / 256; ++r) {
        const int c = r * 256 + t;                       // ascending per thread
        const unsigned int i = (unsigned int)(j * NSAMP + c);
        const unsigned int bits = threefry2x32_x0(0u, 42u, i, i + 0x800000u);
        float u = __uint_as_float((bits >> 9) | 0x3f800000u) - 1.0f;
        u = fmaxf(tiny, u + tiny);
        const float g = -logf(-logf(u));
        const float s = g + logits[c];
        if (s > best || (s == best && c < bi)) { best = s; bi = c; }
    }

    sval[t] = best; sidx[t] = bi;
    __syncthreads();
    for (int stride = 128; stride > 0; stride >>= 1) {
        if (t < stride) {
            const float s2 = sval[t + stride];
            const int   i2 = sidx[t + stride];
            if (s2 > sval[t] || (s2 == sval[t] && i2 < sidx[t])) {
                sval[t] = s2; sidx[t] = i2;
            }
        }
        __syncthreads();
    }
    const int idxj = sidx[0];

    // Gather output column j: out1[b,j] = adj_info[ids[b], idx[j]],
    //                         out2[b,j] = geto_adj_info[geto_ids[b], idx[j]].
    const int b = t;
    out[b * S + j] =
        (float)adj_info[(size_t)ids[b] * MAX_DEG + idxj];
    out[B * S + b * S + j] =
        (float)geto_adj_info[(size_t)geto_ids[b] * MAX_DEG + idxj];
}

// ---------------------------------------------------------------------------
extern "C" void kernel_launch(void* const* d_in, const int* in_sizes, int n_in,
                              void* d_out, int out_size, void* d_ws, size_t ws_size,
                              hipStream_t stream) {
    const int*   adj_info      = (const int*)d_in[0];
    const int*   geto_adj_info = (const int*)d_in[1];
    const float* geto_elms     = (const float*)d_in[2];
    const int*   ids           = (const int*)d_in[3];
    const int*   geto_ids      = (const int*)d_in[4];
    // d_in[5..7]: num_samples / support_size / batch_size (compile-time here)

    float* logits = (float*)d_ws;         // 4096 floats of scratch
    float* out    = (float*)d_out;        // 8192 outputs (2 x 256 x 16)

    geto_logits_kernel<<<32, 256, 0, stream>>>(
        geto_adj_info, geto_elms, ids, geto_ids, logits);

    categorical_gather_kernel<<<16, 256, 0, stream>>>(
        adj_info, geto_adj_info, ids, geto_ids, logits, out);
}